// GAT_58823872086408
// MI455X (gfx1250) — compile-verified
//
#include <hip/hip_runtime.h>
#include <hip/hip_bf16.h>
#include <math.h>

#define N_NODES 50000
#define N_EDGES 800000
#define E_TOT   850000   // N_EDGES + N_NODES self loops
#define N_FEAT  128
#define HDIM    256
#define HOUT    64
#define NGRAPH  128

typedef __attribute__((ext_vector_type(2))) float v2f;
typedef __attribute__((ext_vector_type(8))) float v8f;

static __device__ __forceinline__ void edge_sd(const int* __restrict__ ei, int e, int& s, int& d) {
    if (e < N_EDGES) { s = ei[e]; d = ei[N_EDGES + e]; }
    else             { s = e - N_EDGES; d = s; }
}

// float atomic max via signed/unsigned integer atomics (init to -inf)
static __device__ __forceinline__ void atomicMaxF(float* addr, float v) {
    if (v >= 0.0f) atomicMax((int*)addr, __float_as_int(v));
    else           atomicMin((unsigned int*)addr, __float_as_uint(v));
}

// ---------------------------------------------------------------------------
// WMMA fp32 GEMM: C[M,N] = A[M,K] @ B[K,N], row-major contiguous.
// Requires M%16==0, K%4==0, N%64==0. One wave -> 16x64 tile of C.
// ---------------------------------------------------------------------------
__global__ void wmma_gemm_f32(const float* __restrict__ A, const float* __restrict__ B,
                              float* __restrict__ C, int M, int K, int N) {
    const int lane = threadIdx.x & 31;
    const int wave = threadIdx.x >> 5;
    const int mt   = blockIdx.y * 4 + wave;
    if (mt * 16 >= M) return;                 // wave-uniform guard (EXEC stays all-ones)
    const int row0 = mt * 16;
    const int col0 = blockIdx.x * 64;
    const int m    = lane & 15;
    const int kh   = lane >> 4;

    v8f acc0 = {}, acc1 = {}, acc2 = {}, acc3 = {};
    const float* Arow = A + (size_t)(row0 + m) * K + 2 * kh;

    for (int k0 = 0; k0 < K; k0 += 4) {
        v2f a;
        a.x = Arow[k0];
        a.y = Arow[k0 + 1];
        const float* B0 = B + (size_t)(k0 + 2 * kh) * N + col0 + m;
        const float* B1 = B0 + N;
        v2f b0, b1, b2, b3;
        b0.x = B0[0];  b0.y = B1[0];
        b1.x = B0[16]; b1.y = B1[16];
        b2.x = B0[32]; b2.y = B1[32];
        b3.x = B0[48]; b3.y = B1[48];
        acc0 = __builtin_amdgcn_wmma_f32_16x16x4_f32(false, a, false, b0, (short)0, acc0, false, false);
        acc1 = __builtin_amdgcn_wmma_f32_16x16x4_f32(false, a, false, b1, (short)0, acc1, false, false);
        acc2 = __builtin_amdgcn_wmma_f32_16x16x4_f32(false, a, false, b2, (short)0, acc2, false, false);
        acc3 = __builtin_amdgcn_wmma_f32_16x16x4_f32(false, a, false, b3, (short)0, acc3, false, false);
    }

    float* Crow = C + (size_t)row0 * N + col0 + m;
#pragma unroll
    for (int r = 0; r < 8; ++r) {
        size_t ro = (size_t)(r + 8 * kh) * N;
        Crow[ro +  0] = acc0[r];
        Crow[ro + 16] = acc1[r];
        Crow[ro + 32] = acc2[r];
        Crow[ro + 48] = acc3[r];
    }
}

// ---------------------------------------------------------------------------
// Elementwise / init kernels
// ---------------------------------------------------------------------------
__global__ void fill_f32(float* __restrict__ p, float v, long n) {
    long t = (long)blockIdx.x * blockDim.x + threadIdx.x;
    if (t < n) p[t] = v;
}

__global__ void tanh_k(float* __restrict__ p, long n) {
    long t = (long)blockIdx.x * blockDim.x + threadIdx.x;
    if (t < n) p[t] = tanhf(p[t]);
}

// out[i*C + c] = b[c], C is a power of two
__global__ void init_bias_k(float* __restrict__ outp, const float* __restrict__ b, long n, int Cmask) {
    long t = (long)blockIdx.x * blockDim.x + threadIdx.x;
    if (t < n) outp[t] = b[t & Cmask];
}

// ---------------------------------------------------------------------------
// GAT attention pipeline
// ---------------------------------------------------------------------------
// alpha_s[i] = h[i,:] . a_src ; alpha_d[i] = h[i,:] . a_dst   (one wave per node)
__global__ void alpha_dot_k(const float* __restrict__ h, const float* __restrict__ a_s,
                            const float* __restrict__ a_d, float* __restrict__ os,
                            float* __restrict__ od) {
    int wv   = (int)((blockIdx.x * blockDim.x + threadIdx.x) >> 5);
    int lane = threadIdx.x & 31;
    if (wv >= N_NODES) return;
    const float* hp = h + (size_t)wv * HDIM;
    float s = 0.f, d = 0.f;
#pragma unroll
    for (int j = 0; j < 8; ++j) {
        int c = lane + 32 * j;
        float v = hp[c];
        s += v * a_s[c];
        d += v * a_d[c];
    }
    for (int off = 16; off; off >>= 1) {
        s += __shfl_down(s, off, 32);
        d += __shfl_down(d, off, 32);
    }
    if (lane == 0) { os[wv] = s; od[wv] = d; }
}

__global__ void edge_max_k(const int* __restrict__ ei, const float* __restrict__ as,
                           const float* __restrict__ ad, float* __restrict__ m) {
    int e = blockIdx.x * blockDim.x + threadIdx.x;
    if (e >= E_TOT) return;
    int s, d; edge_sd(ei, e, s, d);
    float v = as[s] + ad[d];
    v = v > 0.f ? v : 0.2f * v;
    atomicMaxF(&m[d], v);
}

__global__ void edge_expsum_k(const int* __restrict__ ei, const float* __restrict__ as,
                              const float* __restrict__ ad, const float* __restrict__ m,
                              float* __restrict__ w, float* __restrict__ z) {
    int e = blockIdx.x * blockDim.x + threadIdx.x;
    if (e >= E_TOT) return;
    int s, d; edge_sd(ei, e, s, d);
    float v = as[s] + ad[d];
    v = v > 0.f ? v : 0.2f * v;
    float ex = __expf(v - m[d]);
    w[e] = ex;
    atomicAdd(&z[d], ex);
}

__global__ void edge_div_k(const int* __restrict__ ei, float* __restrict__ w,
                           const float* __restrict__ z) {
    int e = blockIdx.x * blockDim.x + threadIdx.x;
    if (e >= E_TOT) return;
    int s, d; edge_sd(ei, e, s, d);
    w[e] = w[e] / z[d];   // z >= exp(0) contribution from self loop => nonzero
}

// out[dst] += alpha * h[src], 4 channels per thread (float4 gather + 4 f32 atomics)
__global__ void edge_scatter_k(const int* __restrict__ ei, const float* __restrict__ w,
                               const float* __restrict__ hpre, float* __restrict__ outp) {
    long t = (long)blockIdx.x * blockDim.x + threadIdx.x;
    if (t >= (long)E_TOT * (HDIM / 4)) return;
    int e = (int)(t >> 6);            // HDIM/4 == 64 channel groups
    int c = (int)(t & 63) << 2;
    int s, d; edge_sd(ei, e, s, d);
    float a = w[e];
    const float4 v = *(const float4*)(hpre + (size_t)s * HDIM + c);
    float* o = outp + (size_t)d * HDIM + c;
    atomicAdd(o + 0, a * v.x);
    atomicAdd(o + 1, a * v.y);
    atomicAdd(o + 2, a * v.z);
    atomicAdd(o + 3, a * v.w);
}

// ---------------------------------------------------------------------------
// Global pooling
// ---------------------------------------------------------------------------
__global__ void pool_max_sum_k(const float* __restrict__ h, const int* __restrict__ batch,
                               float* __restrict__ gmax, float* __restrict__ gmaxn,
                               float* __restrict__ gsum) {
    long t = (long)blockIdx.x * blockDim.x + threadIdx.x;
    if (t >= (long)N_NODES * HDIM) return;
    int i = (int)(t >> 8);
    int c = (int)(t & 255);
    int g = batch[i];
    float v = h[t];
    atomicMaxF(&gmax[(size_t)g * HDIM + c], v);
    atomicMaxF(&gmaxn[(size_t)g * HDIM + c], -v);
    atomicAdd(&gsum[(size_t)g * HDIM + c], v);
}

__global__ void pool_cnt_k(const int* __restrict__ batch, float* __restrict__ cnt) {
    int i = blockIdx.x * blockDim.x + threadIdx.x;
    if (i < N_NODES) atomicAdd(&cnt[batch[i]], 1.0f);
}

// pooled[g, 0:256]=gmax  [256:512]=gmaxneg  [512:768]=gsum/max(cnt,1)
__global__ void pool_final_k(const float* __restrict__ gmax, const float* __restrict__ gmaxn,
                             const float* __restrict__ gsum, const float* __restrict__ cnt,
                             float* __restrict__ pooled) {
    int t = blockIdx.x * blockDim.x + threadIdx.x;
    if (t >= NGRAPH * HDIM) return;
    int g = t >> 8;
    int c = t & 255;
    float* pr = pooled + (size_t)g * (3 * HDIM);
    pr[c]            = gmax[t];
    pr[HDIM + c]     = gmaxn[t];
    pr[2 * HDIM + c] = gsum[t] / fmaxf(cnt[g], 1.0f);
}

// hg[i,c] += pg[batch[i],c]
__global__ void add_pg_k(float* __restrict__ hg, const float* __restrict__ pg,
                         const int* __restrict__ batch) {
    long t = (long)blockIdx.x * blockDim.x + threadIdx.x;
    if (t >= (long)N_NODES * HOUT) return;
    int i = (int)(t >> 6);
    int c = (int)(t & 63);
    hg[t] += pg[(size_t)batch[i] * HOUT + c];
}

// ---------------------------------------------------------------------------
// GCN
// ---------------------------------------------------------------------------
__global__ void edge_deg_k(const int* __restrict__ ei, float* __restrict__ deg) {
    int e = blockIdx.x * blockDim.x + threadIdx.x;
    if (e >= E_TOT) return;
    int s, d; edge_sd(ei, e, s, d);
    atomicAdd(&deg[d], 1.0f);
}

__global__ void dinv_k(const float* __restrict__ deg, float* __restrict__ dinv) {
    int i = blockIdx.x * blockDim.x + threadIdx.x;
    if (i < N_NODES) dinv[i] = rsqrtf(fmaxf(deg[i], 1.0f));
}

__global__ void gcn_scatter_k(const int* __restrict__ ei, const float* __restrict__ dinv,
                              const float* __restrict__ hg, float* __restrict__ outp) {
    long t = (long)blockIdx.x * blockDim.x + threadIdx.x;
    if (t >= (long)E_TOT * (HOUT / 4)) return;
    int e = (int)(t >> 4);            // HOUT/4 == 16 channel groups
    int c = (int)(t & 15) << 2;
    int s, d; edge_sd(ei, e, s, d);
    float nrm = dinv[s] * dinv[d];
    const float4 v = *(const float4*)(hg + (size_t)s * HOUT + c);
    float* o = outp + (size_t)d * HOUT + c;
    atomicAdd(o + 0, nrm * v.x);
    atomicAdd(o + 1, nrm * v.y);
    atomicAdd(o + 2, nrm * v.z);
    atomicAdd(o + 3, nrm * v.w);
}

// out[i] = h_final[i,:] . Wo + bo  (one wave per node, 64 dims)
__global__ void out_dot_k(const float* __restrict__ hf, const float* __restrict__ Wo,
                          const float* __restrict__ bo, float* __restrict__ out) {
    int wv   = (int)((blockIdx.x * blockDim.x + threadIdx.x) >> 5);
    int lane = threadIdx.x & 31;
    if (wv >= N_NODES) return;
    const float* hp = hf + (size_t)wv * HOUT;
    float s = hp[lane] * Wo[lane] + hp[lane + 32] * Wo[lane + 32];
    for (int off = 16; off; off >>= 1) s += __shfl_down(s, off, 32);
    if (lane == 0) out[wv] = s + bo[0];
}

// ---------------------------------------------------------------------------
// Host orchestration
// ---------------------------------------------------------------------------
static inline int cdiv(long a, long b) { return (int)((a + b - 1) / b); }

extern "C" void kernel_launch(void* const* d_in, const int* in_sizes, int n_in,
                              void* d_out, int out_size, void* d_ws, size_t ws_size,
                              hipStream_t stream) {
    const float* x   = (const float*)d_in[0];
    const float* W1  = (const float*)d_in[1];
    const float* as1 = (const float*)d_in[2];
    const float* ad1 = (const float*)d_in[3];
    const float* b1  = (const float*)d_in[4];
    const float* W2  = (const float*)d_in[5];
    const float* as2 = (const float*)d_in[6];
    const float* ad2 = (const float*)d_in[7];
    const float* b2  = (const float*)d_in[8];
    const float* W3  = (const float*)d_in[9];
    const float* as3 = (const float*)d_in[10];
    const float* ad3 = (const float*)d_in[11];
    const float* b3  = (const float*)d_in[12];
    const float* Wg  = (const float*)d_in[13];
    const float* bg  = (const float*)d_in[14];
    const float* Wo  = (const float*)d_in[15];
    const float* bo  = (const float*)d_in[16];
    const int*   ei  = (const int*)d_in[17];
    const int*   bat = (const int*)d_in[18];

    float* ws     = (float*)d_ws;
    float* bufA   = ws;                               // [N,256] (later reused for hg [N,64])
    float* bufB   = bufA  + (size_t)N_NODES * HDIM;   // [N,256]
    float* alphas = bufB  + (size_t)N_NODES * HDIM;   // [N]
    float* alphad = alphas + N_NODES;
    float* mbuf   = alphad + N_NODES;
    float* zbuf   = mbuf   + N_NODES;
    float* deg    = zbuf   + N_NODES;
    float* dinv   = deg    + N_NODES;
    float* wbuf   = dinv   + N_NODES;                 // [E_TOT]
    float* gmax   = wbuf   + E_TOT;                   // [G,256]
    float* gmaxn  = gmax   + (size_t)NGRAPH * HDIM;
    float* gsum   = gmaxn  + (size_t)NGRAPH * HDIM;
    float* cnt    = gsum   + (size_t)NGRAPH * HDIM;   // [G]
    float* pooled = cnt    + NGRAPH;                  // [G,768]
    float* pg     = pooled + (size_t)NGRAPH * 3 * HDIM; // [G,64]

    float* out_pred = (float*)d_out;                  // [N,1]
    float* hfin     = out_pred + N_NODES;             // [N,64] -> second tuple output

    const int TB = 256;
    const long NH  = (long)N_NODES * HDIM;
    const long NHO = (long)N_NODES * HOUT;
    const int  EB  = cdiv(E_TOT, TB);

    auto gemm = [&](const float* A, const float* B, float* C, int M, int K, int N) {
        dim3 g(N / 64, cdiv(M / 16, 4));
        wmma_gemm_f32<<<g, dim3(128), 0, stream>>>(A, B, C, M, K, N);
    };

    auto gat_layer = [&](const float* in_feat, int in_dim, const float* Wl,
                         const float* asv, const float* adv, const float* bl) {
        // h_pre = in_feat @ Wl -> bufA  (WMMA)
        gemm(in_feat, Wl, bufA, N_NODES, in_dim, HDIM);
        // per-node attention logits
        alpha_dot_k<<<cdiv((long)N_NODES * 32, TB), TB, 0, stream>>>(bufA, asv, adv, alphas, alphad);
        // segment softmax over incoming edges
        fill_f32<<<cdiv(N_NODES, TB), TB, 0, stream>>>(mbuf, -INFINITY, N_NODES);
        fill_f32<<<cdiv(N_NODES, TB), TB, 0, stream>>>(zbuf, 0.0f, N_NODES);
        edge_max_k<<<EB, TB, 0, stream>>>(ei, alphas, alphad, mbuf);
        edge_expsum_k<<<EB, TB, 0, stream>>>(ei, alphas, alphad, mbuf, wbuf, zbuf);
        edge_div_k<<<EB, TB, 0, stream>>>(ei, wbuf, zbuf);
        // out = sum_e alpha_e * h_pre[src] + b  -> bufB, then tanh
        init_bias_k<<<cdiv(NH, TB), TB, 0, stream>>>(bufB, bl, NH, HDIM - 1);
        edge_scatter_k<<<cdiv((long)E_TOT * (HDIM / 4), TB), TB, 0, stream>>>(ei, wbuf, bufA, bufB);
        tanh_k<<<cdiv(NH, TB), TB, 0, stream>>>(bufB, NH);
    };

    // --- 3 GAT layers ---
    gat_layer(x,    N_FEAT, W1, as1, ad1, b1);
    gat_layer(bufB, HDIM,   W2, as2, ad2, b2);
    gat_layer(bufB, HDIM,   W3, as3, ad3, b3);
    // bufB now holds h3 = tanh(gat3)  [N,256]

    // --- global pooling ---
    fill_f32<<<cdiv((long)NGRAPH * HDIM, TB), TB, 0, stream>>>(gmax,  -INFINITY, (long)NGRAPH * HDIM);
    fill_f32<<<cdiv((long)NGRAPH * HDIM, TB), TB, 0, stream>>>(gmaxn, -INFINITY, (long)NGRAPH * HDIM);
    fill_f32<<<cdiv((long)NGRAPH * HDIM, TB), TB, 0, stream>>>(gsum,  0.0f,      (long)NGRAPH * HDIM);
    fill_f32<<<cdiv(NGRAPH, TB), TB, 0, stream>>>(cnt, 0.0f, NGRAPH);
    pool_max_sum_k<<<cdiv(NH, TB), TB, 0, stream>>>(bufB, bat, gmax, gmaxn, gsum);
    pool_cnt_k<<<cdiv(N_NODES, TB), TB, 0, stream>>>(bat, cnt);
    pool_final_k<<<cdiv((long)NGRAPH * HDIM, TB), TB, 0, stream>>>(gmax, gmaxn, gsum, cnt, pooled);

    // --- GCN, split: concat(h,pooled[batch]) @ Wg == h@Wg[0:256] + (pooled@Wg[256:])[batch]
    gemm(bufB,   Wg,              bufA, N_NODES, HDIM,     HOUT);   // hg -> bufA (reuse)
    gemm(pooled, Wg + 256 * HOUT, pg,   NGRAPH,  3 * HDIM, HOUT);   // pg [128,64]
    add_pg_k<<<cdiv(NHO, TB), TB, 0, stream>>>(bufA, pg, bat);

    fill_f32<<<cdiv(N_NODES, TB), TB, 0, stream>>>(deg, 0.0f, N_NODES);
    edge_deg_k<<<EB, TB, 0, stream>>>(ei, deg);
    dinv_k<<<cdiv(N_NODES, TB), TB, 0, stream>>>(deg, dinv);

    init_bias_k<<<cdiv(NHO, TB), TB, 0, stream>>>(hfin, bg, NHO, HOUT - 1);
    gcn_scatter_k<<<cdiv((long)E_TOT * (HOUT / 4), TB), TB, 0, stream>>>(ei, dinv, bufA, hfin);
    tanh_k<<<cdiv(NHO, TB), TB, 0, stream>>>(hfin, NHO);

    // --- final linear: out = h_final @ Wo + bo ---
    out_dot_k<<<cdiv((long)N_NODES * 32, TB), TB, 0, stream>>>(hfin, Wo, bo, out_pred);

    (void)in_sizes; (void)n_in; (void)out_size; (void)ws_size;
}